// EnhancedGCN_20435454395157
// MI455X (gfx1250) — compile-verified
//
#include <hip/hip_runtime.h>
#include <hip/hip_bf16.h>

#define BN_EPS 1e-5f

typedef __attribute__((ext_vector_type(2))) float v2f;
typedef __attribute__((ext_vector_type(8))) float v8f;

// ---------------- utility ----------------
__global__ void fill_kernel(float* __restrict__ p, float v, int n) {
    int i = blockIdx.x * blockDim.x + threadIdx.x;
    for (; i < n; i += gridDim.x * blockDim.x) p[i] = v;
}

__global__ void deg_kernel(const int* __restrict__ dst, float* __restrict__ deg, int nE) {
    int e = blockIdx.x * blockDim.x + threadIdx.x;
    if (e < nE) atomicAdd(&deg[dst[e]], 1.0f);
}

__global__ void dinv_kernel(float* __restrict__ p, int n) {
    int i = blockIdx.x * blockDim.x + threadIdx.x;
    if (i < n) p[i] = rsqrtf(p[i] + 1.0f);
}

// ---------------- WMMA GEMM: Y[N x HOUT] = X[N x 128] @ W[128 x HOUT] ----------------
// 256 threads = 8 waves; each wave owns a 16-row stripe; block covers 128 rows.
// W is staged in LDS K-pair-interleaved: pair index kp=k/2 stores (W[k][n], W[k+1][n])
// contiguously, so each B fragment is one aligned ds_load_b64 (no repack movs).
template <int HOUT>
__global__ __launch_bounds__(256) void gemm_wmma_kernel(const float* __restrict__ X,
                                                        const float* __restrict__ W,
                                                        float* __restrict__ Y, int n) {
    extern __shared__ float Ws[];  // 128*HOUT floats, pair-interleaved
    const int tid = threadIdx.x;
    for (int i = tid; i < 128 * HOUT; i += 256) {
        const int k = i / HOUT;
        const int c = i - k * HOUT;
        Ws[((k >> 1) * HOUT + c) * 2 + (k & 1)] = W[i];
    }
    __syncthreads();

    const int wave = tid >> 5;
    const int lane = tid & 31;
    const int m    = lane & 15;   // M for A, N-col for B/C/D
    const int half = lane >> 4;   // 0 or 1

    const int arow = blockIdx.x * 128 + wave * 16 + m;  // A-matrix row for this lane
    const int rr   = arow < n ? arow : 0;               // clamp: garbage rows never stored
    const float* xr = X + (size_t)rr * 128;
    const v2f* wp  = (const v2f*)Ws;

    v8f acc[HOUT / 16];
#pragma unroll
    for (int nt = 0; nt < HOUT / 16; ++nt) acc[nt] = v8f{0.f, 0.f, 0.f, 0.f, 0.f, 0.f, 0.f, 0.f};

    for (int kt = 0; kt < 32; ++kt) {
        const int kp = kt * 2 + half;  // K-pair index; covers K = 2*kp, 2*kp+1
        v2f a = *(const v2f*)(xr + 2 * kp);
        const v2f* wrow = wp + kp * HOUT;
#pragma unroll
        for (int nt = 0; nt < HOUT / 16; ++nt) {
            v2f b = wrow[nt * 16 + m];
            acc[nt] = __builtin_amdgcn_wmma_f32_16x16x4_f32(
                /*neg_a=*/false, a, /*neg_b=*/false, b,
                /*c_mod=*/(short)0, acc[nt], /*reuse_a=*/false, /*reuse_b=*/false);
        }
    }

    // C/D layout: VGPR p, lane -> M = p + 8*half, N = m
    const int rbase = blockIdx.x * 128 + wave * 16 + 8 * half;
#pragma unroll
    for (int nt = 0; nt < HOUT / 16; ++nt) {
#pragma unroll
        for (int p = 0; p < 8; ++p) {
            const int r = rbase + p;
            if (r < n) Y[(size_t)r * HOUT + nt * 16 + m] = acc[nt][p];
        }
    }
}

// ---------------- agg init: agg[i,f] = h[i,f]*dinv[i]^2 + bias[f]  (self-loop + bias) ----
template <int F>
__global__ void init_agg_kernel(const float* __restrict__ h, const float* __restrict__ dinv,
                                const float* __restrict__ bias, float* __restrict__ agg, int n) {
    constexpr int PF = F / 4;
    int idx = blockIdx.x * blockDim.x + threadIdx.x;
    if (idx >= n * PF) return;
    int i = idx / PF;
    int j = idx - i * PF;
    float di = dinv[i];
    float c  = di * di;
    float4 hv = ((const float4*)h)[idx];
    float4 bv = ((const float4*)bias)[j];
    float4 o;
    o.x = hv.x * c + bv.x;
    o.y = hv.y * c + bv.y;
    o.z = hv.z * c + bv.z;
    o.w = hv.w * c + bv.w;
    ((float4*)agg)[idx] = o;
}

// ---------------- edge scatter: agg[dst] += h[src] * dinv[src]*dinv[dst] ----------------
// One wave per edge; edge index forced wave-uniform via readfirstlane so the
// src/dst/dinv loads become scalar and the vector pipe only does gather+atomics.
template <int F>
__global__ __launch_bounds__(256) void scatter_kernel(const int* __restrict__ src,
                                                      const int* __restrict__ dst,
                                                      const float* __restrict__ h,
                                                      const float* __restrict__ dinv,
                                                      float* __restrict__ agg, int nE) {
    constexpr int VPT = F / 32;  // floats per lane (4 for F=128, 2 for F=64)
    int e = __builtin_amdgcn_readfirstlane((int)(blockIdx.x * 8 + (threadIdx.x >> 5)));
    if (e >= nE) return;  // wave-uniform exit
    const int s = src[e];
    const int d = dst[e];
    const float coef = dinv[s] * dinv[d];
    const int lane = threadIdx.x & 31;
    const float* hp = h + (size_t)s * F + lane * VPT;
    float* ap = agg + (size_t)d * F + lane * VPT;
    if constexpr (VPT == 4) {
        float4 hv = *(const float4*)hp;
        atomicAdd(ap + 0, hv.x * coef);
        atomicAdd(ap + 1, hv.y * coef);
        atomicAdd(ap + 2, hv.z * coef);
        atomicAdd(ap + 3, hv.w * coef);
    } else {
        float2 hv = *(const float2*)hp;
        atomicAdd(ap + 0, hv.x * coef);
        atomicAdd(ap + 1, hv.y * coef);
    }
}

// ---------------- column stats over z[N x 128]: stats[0:128]=sum, stats[128:256]=sumsq ---
__global__ __launch_bounds__(128) void stats_kernel(const float* __restrict__ z,
                                                    float* __restrict__ stats, int n) {
    const int f = threadIdx.x;  // 128 threads = 128 features
    float s = 0.f, q = 0.f;
    for (int r = blockIdx.x; r < n; r += gridDim.x) {
        float v = z[(size_t)r * 128 + f];
        s += v;
        q += v * v;
    }
    atomicAdd(&stats[f], s);
    atomicAdd(&stats[128 + f], q);
}

// ---------------- BN (training stats) + ReLU (+ optional skip), float4 vectorized -------
__global__ void bn_relu_kernel(const float* __restrict__ z, const float* __restrict__ stats,
                               const float* __restrict__ g, const float* __restrict__ be,
                               const float* __restrict__ skip, float* __restrict__ out,
                               int n, float invN) {
    int idx = blockIdx.x * blockDim.x + threadIdx.x;
    if (idx >= n * 32) return;
    int j = idx & 31;  // float4 group within the 128 features
    float4 sv = ((const float4*)stats)[j];
    float4 qv = ((const float4*)(stats + 128))[j];
    float4 gv = ((const float4*)g)[j];
    float4 bv = ((const float4*)be)[j];
    float4 zv = ((const float4*)z)[idx];
    float4 o;
    {
        float mm = sv.x * invN; float var = qv.x * invN - mm * mm;
        float sc = gv.x * rsqrtf(var + BN_EPS);
        o.x = fmaxf((zv.x - mm) * sc + bv.x, 0.f);
    }
    {
        float mm = sv.y * invN; float var = qv.y * invN - mm * mm;
        float sc = gv.y * rsqrtf(var + BN_EPS);
        o.y = fmaxf((zv.y - mm) * sc + bv.y, 0.f);
    }
    {
        float mm = sv.z * invN; float var = qv.z * invN - mm * mm;
        float sc = gv.z * rsqrtf(var + BN_EPS);
        o.z = fmaxf((zv.z - mm) * sc + bv.z, 0.f);
    }
    {
        float mm = sv.w * invN; float var = qv.w * invN - mm * mm;
        float sc = gv.w * rsqrtf(var + BN_EPS);
        o.w = fmaxf((zv.w - mm) * sc + bv.w, 0.f);
    }
    if (skip) {
        float4 kv = ((const float4*)skip)[idx];
        o.x += kv.x; o.y += kv.y; o.z += kv.z; o.w += kv.w;
    }
    ((float4*)out)[idx] = o;
}

// ---------------- launch ----------------
extern "C" void kernel_launch(void* const* d_in, const int* in_sizes, int n_in,
                              void* d_out, int out_size, void* d_ws, size_t ws_size,
                              hipStream_t stream) {
    (void)n_in; (void)out_size; (void)ws_size;
    const float* x   = (const float*)d_in[0];
    const int*   ei  = (const int*)d_in[1];
    const float* W1  = (const float*)d_in[2];
    const float* b1  = (const float*)d_in[3];
    const float* g1  = (const float*)d_in[4];
    const float* be1 = (const float*)d_in[5];
    const float* W2  = (const float*)d_in[6];
    const float* b2  = (const float*)d_in[7];
    const float* g2  = (const float*)d_in[8];
    const float* be2 = (const float*)d_in[9];
    const float* W3  = (const float*)d_in[10];
    const float* b3  = (const float*)d_in[11];
    float* out = (float*)d_out;

    const int N = in_sizes[0] / 128;
    const int E = in_sizes[1] / 2;
    const int* srcI = ei;
    const int* dstI = ei + E;

    float* ws = (float*)d_ws;
    auto alignup = [](size_t v) { return (v + 255) & ~(size_t)255; };
    size_t off = 0;
    float* p_dinv = ws + off; off = alignup(off + (size_t)N);
    float* p_h    = ws + off; off = alignup(off + (size_t)N * 128);
    float* p_a    = ws + off; off = alignup(off + (size_t)N * 128);
    float* p_x1   = ws + off; off = alignup(off + (size_t)N * 128);
    float* p_st   = ws + off; off = alignup(off + 256);

    const int TB = 256;
    const float invN = 1.0f / (float)N;
    const unsigned gN   = (unsigned)((N + TB - 1) / TB);
    const unsigned gE   = (unsigned)((E + TB - 1) / TB);
    const unsigned gMM  = (unsigned)((N + 127) / 128);
    const unsigned gEl  = (unsigned)(((size_t)N * 32 + TB - 1) / TB);  // N*128/4 threads
    const unsigned gEl3 = (unsigned)(((size_t)N * 16 + TB - 1) / TB);  // N*64/4 threads
    const unsigned gSc  = (unsigned)((E + 7) / 8);                     // 8 edges (waves) per block

    // degrees -> dinv (in place)
    fill_kernel<<<gN, TB, 0, stream>>>(p_dinv, 0.f, N);
    deg_kernel<<<gE, TB, 0, stream>>>(dstI, p_dinv, E);
    dinv_kernel<<<gN, TB, 0, stream>>>(p_dinv, N);

    // ---- layer 1 ----
    gemm_wmma_kernel<128><<<gMM, TB, 128 * 128 * 4, stream>>>(x, W1, p_h, N);
    init_agg_kernel<128><<<gEl, TB, 0, stream>>>(p_h, p_dinv, b1, p_a, N);
    scatter_kernel<128><<<gSc, TB, 0, stream>>>(srcI, dstI, p_h, p_dinv, p_a, E);
    fill_kernel<<<1, TB, 0, stream>>>(p_st, 0.f, 256);
    stats_kernel<<<1024, 128, 0, stream>>>(p_a, p_st, N);
    bn_relu_kernel<<<gEl, TB, 0, stream>>>(p_a, p_st, g1, be1, nullptr, p_x1, N, invN);

    // ---- layer 2 (skip = x1; BN output written in place into p_a, which becomes x2) ----
    gemm_wmma_kernel<128><<<gMM, TB, 128 * 128 * 4, stream>>>(p_x1, W2, p_h, N);
    init_agg_kernel<128><<<gEl, TB, 0, stream>>>(p_h, p_dinv, b2, p_a, N);
    scatter_kernel<128><<<gSc, TB, 0, stream>>>(srcI, dstI, p_h, p_dinv, p_a, E);
    fill_kernel<<<1, TB, 0, stream>>>(p_st, 0.f, 256);
    stats_kernel<<<1024, 128, 0, stream>>>(p_a, p_st, N);
    bn_relu_kernel<<<gEl, TB, 0, stream>>>(p_a, p_st, g2, be2, p_x1, p_a, N, invN);

    // ---- layer 3 (no BN/ReLU; result goes straight to d_out) ----
    gemm_wmma_kernel<64><<<gMM, TB, 128 * 64 * 4, stream>>>(p_a, W3, p_h, N);
    init_agg_kernel<64><<<gEl3, TB, 0, stream>>>(p_h, p_dinv, b3, out, N);
    scatter_kernel<64><<<gSc, TB, 0, stream>>>(srcI, dstI, p_h, p_dinv, out, E);
}